// GINConv_63642825392373
// MI455X (gfx1250) — compile-verified
//
#include <hip/hip_runtime.h>

typedef __attribute__((ext_vector_type(2))) float v2f;
typedef __attribute__((ext_vector_type(8))) float v8f;

#define NN 2048
#define FF 32
#define EE 16
#define KK 32

// ---------------- Phase A: per-node preprocessing ----------------
// npb[n,f]   = (nodes @ W_ne[:F])[n,f] + b_ne[f]      (bias folded in)
// nterm[n,f] = relu((nodes @ W_n)[n,f] + b_n[f])
__global__ __launch_bounds__(256) void gin_preproc(
    const float* __restrict__ nodes,
    const float* __restrict__ W_ne,
    const float* __restrict__ b_ne,
    const float* __restrict__ W_n,
    const float* __restrict__ b_n,
    float* __restrict__ npb,
    float* __restrict__ nterm)
{
    const int n = blockIdx.x * 256 + threadIdx.x;
    if (n >= NN) return;
    float x[FF];
#pragma unroll
    for (int k = 0; k < FF; ++k) x[k] = nodes[n * FF + k];
#pragma unroll 4
    for (int f = 0; f < FF; ++f) {
        float a = b_ne[f];
        float b = b_n[f];
#pragma unroll
        for (int k = 0; k < FF; ++k) {
            a = fmaf(x[k], W_ne[k * FF + f], a);
            b = fmaf(x[k], W_n[k * FF + f], b);
        }
        npb[n * FF + f]   = a;
        nterm[n * FF + f] = fmaxf(b, 0.0f);
    }
}

// ---------------- Phase B: one block per row i ----------------
// Fuses: edges copy, edge_part WMMA GEMM, relu+mask+sum (msg), and the
// final (1+eps)*nterm + msg -> @W_net -> relu row epilogue.
__global__ __launch_bounds__(256) void gin_row(
    const float* __restrict__ adj,      // [N,N]
    const float* __restrict__ edges,    // [N,N,E]
    const float* __restrict__ W_ne,     // [F+E,F] (rows F.. used here)
    const float* __restrict__ W_net,    // [F,K]
    const float* __restrict__ b_net,    // [K]
    const float* __restrict__ eps,      // [1]
    const float* __restrict__ npb,      // [N,F]
    const float* __restrict__ nterm,    // [N,F]
    float* __restrict__ out_adj,        // [N,N]
    float* __restrict__ out_edges,      // [N,N,E]
    float* __restrict__ out_out)        // [N,K]
{
    __shared__ float adj_s[NN];
    __shared__ float msg_s[FF];
    __shared__ float v_s[FF];

    const int i   = blockIdx.x;
    const int tid = threadIdx.x;

    // Stage adj row into LDS and emit the adj copy.
    for (int c = tid; c < NN; c += 256) {
        float v = adj[(size_t)i * NN + c];
        adj_s[c] = v;
        out_adj[(size_t)i * NN + c] = v;
    }
    if (tid < FF) msg_s[tid] = 0.0f;

    const int lane = tid & 31;
    const int wave = tid >> 5;
    const int m    = lane & 15;   // tile row (M) / output col (N) index
    const int hi   = lane >> 4;   // lane half selects K pairs / M upper half

    // B fragments for V_WMMA_F32_16X16X4_F32: B is 4x16 (K x N) per K-step.
    // Layout: VGPR0 = rows K0 (lanes 0-15) / K2 (lanes 16-31); VGPR1 = K1/K3.
    v2f Bf[4][2];
#pragma unroll
    for (int kk = 0; kk < 4; ++kk)
#pragma unroll
        for (int t = 0; t < 2; ++t) {
            const int e0 = 4 * kk + 2 * hi;
            const int f_ = 16 * t + m;
            Bf[kk][t].x = W_ne[(FF + e0)     * FF + f_];
            Bf[kk][t].y = W_ne[(FF + e0 + 1) * FF + f_];
        }

    __syncthreads();

    float msg0 = 0.0f, msg1 = 0.0f;

    // 128 tiles of 16 neighbors each, round-robined over 8 waves.
    for (int tile = wave; tile < NN / 16; tile += 8) {
        const int j0 = tile * 16;
        const int jm = j0 + m;
        const float* erow = edges     + ((size_t)i * NN + jm) * EE;
        float*       orow = out_edges + ((size_t)i * NN + jm) * EE;

        // A fragments (16x4 f32): lane<16 holds K={4kk,4kk+1}, lane>=16 K={4kk+2,4kk+3}.
        v2f a0 = *(const v2f*)(erow + 0  + 2 * hi);
        v2f a1 = *(const v2f*)(erow + 4  + 2 * hi);
        v2f a2 = *(const v2f*)(erow + 8  + 2 * hi);
        v2f a3 = *(const v2f*)(erow + 12 + 2 * hi);

        // Fused edges copy (same registers, read-once stream).
        *(v2f*)(orow + 0  + 2 * hi) = a0;
        *(v2f*)(orow + 4  + 2 * hi) = a1;
        *(v2f*)(orow + 8  + 2 * hi) = a2;
        *(v2f*)(orow + 12 + 2 * hi) = a3;

        // adj values for this lane's 8 accumulator rows (j = j0 + r + 8*hi).
        float av[8];
        float amax = 0.0f;
#pragma unroll
        for (int r = 0; r < 8; ++r) {
            av[r] = adj_s[j0 + r + 8 * hi];
            amax = fmaxf(amax, fabsf(av[r]));
        }
        // ~44% of tiles have all-zero adj -> skip the GEMM (uniform wave branch,
        // EXEC stays all-ones for WMMA).
        if (__ballot(amax > 0.0f) == 0ull) continue;

        v8f acc0 = {}; v8f acc1 = {};
        acc0 = __builtin_amdgcn_wmma_f32_16x16x4_f32(false, a0, false, Bf[0][0], (short)0, acc0, false, false);
        acc1 = __builtin_amdgcn_wmma_f32_16x16x4_f32(false, a0, false, Bf[0][1], (short)0, acc1, false, false);
        acc0 = __builtin_amdgcn_wmma_f32_16x16x4_f32(false, a1, false, Bf[1][0], (short)0, acc0, false, false);
        acc1 = __builtin_amdgcn_wmma_f32_16x16x4_f32(false, a1, false, Bf[1][1], (short)0, acc1, false, false);
        acc0 = __builtin_amdgcn_wmma_f32_16x16x4_f32(false, a2, false, Bf[2][0], (short)0, acc0, false, false);
        acc1 = __builtin_amdgcn_wmma_f32_16x16x4_f32(false, a2, false, Bf[2][1], (short)0, acc1, false, false);
        acc0 = __builtin_amdgcn_wmma_f32_16x16x4_f32(false, a3, false, Bf[3][0], (short)0, acc0, false, false);
        acc1 = __builtin_amdgcn_wmma_f32_16x16x4_f32(false, a3, false, Bf[3][1], (short)0, acc1, false, false);

        // ne = relu(edge_part + node_part[j] + b_ne); msg += adj * ne.
        // C/D layout: VGPR r -> M = r + 8*hi, N = lane%16 (+16 for acc1).
#pragma unroll
        for (int r = 0; r < 8; ++r) {
            const int j = j0 + r + 8 * hi;
            const float e0 = acc0[r] + npb[j * FF + m];
            const float e1 = acc1[r] + npb[j * FF + 16 + m];
            msg0 = fmaf(av[r], fmaxf(e0, 0.0f), msg0);
            msg1 = fmaf(av[r], fmaxf(e1, 0.0f), msg1);
        }
    }

    atomicAdd(&msg_s[m],      msg0);   // ds_add_f32
    atomicAdd(&msg_s[16 + m], msg1);
    __syncthreads();

    // Row epilogue: out[i] = relu(((1+eps)*nterm[i] + msg) @ W_net + b_net)
    if (tid < FF)
        v_s[tid] = fmaf(1.0f + eps[0], nterm[i * FF + tid], msg_s[tid]);
    __syncthreads();
    if (tid < KK) {
        float acc = b_net[tid];
#pragma unroll
        for (int f = 0; f < FF; ++f)
            acc = fmaf(v_s[f], W_net[f * KK + tid], acc);
        out_out[i * KK + tid] = fmaxf(acc, 0.0f);
    }
}

extern "C" void kernel_launch(void* const* d_in, const int* in_sizes, int n_in,
                              void* d_out, int out_size, void* d_ws, size_t ws_size,
                              hipStream_t stream) {
    (void)in_sizes; (void)n_in; (void)out_size; (void)ws_size;
    const float* adj    = (const float*)d_in[0];
    const float* nodes  = (const float*)d_in[1];
    const float* edges  = (const float*)d_in[2];
    const float* W_ne   = (const float*)d_in[3];
    const float* b_ne   = (const float*)d_in[4];
    const float* W_n    = (const float*)d_in[5];
    const float* b_n    = (const float*)d_in[6];
    const float* W_net  = (const float*)d_in[7];
    const float* b_net  = (const float*)d_in[8];
    const float* eps    = (const float*)d_in[9];

    // Output tuple (adj, out, edges) flattened in order.
    float* out_adj   = (float*)d_out;
    float* out_out   = out_adj + (size_t)NN * NN;
    float* out_edges = out_out + (size_t)NN * KK;

    // Workspace: npb [N,F] then nterm [N,F].
    float* npb   = (float*)d_ws;
    float* nterm = npb + (size_t)NN * FF;

    gin_preproc<<<NN / 256, 256, 0, stream>>>(nodes, W_ne, b_ne, W_n, b_n, npb, nterm);
    gin_row<<<NN, 256, 0, stream>>>(adj, edges, W_ne, W_net, b_net, eps,
                                    npb, nterm, out_adj, out_edges, out_out);
}